// GlobalAttention_76227079569843
// MI455X (gfx1250) — compile-verified
//
#include <hip/hip_runtime.h>
#include <hip/hip_bf16.h>

// ---------------------------------------------------------------------------
// Shapes (fixed by the reference): B=16, T=64, S=256, D=512
// ---------------------------------------------------------------------------
#define BB 16
#define TT 64
#define SS 256
#define DD 512

typedef __attribute__((ext_vector_type(2))) float v2f;
typedef __attribute__((ext_vector_type(8))) float v8f;

__device__ __forceinline__ v8f wmma_f32_k4(v2f a, v2f b, v8f c) {
  // D = A(16x4 f32) * B(4x16 f32) + C(16x16 f32)
  return __builtin_amdgcn_wmma_f32_16x16x4_f32(
      /*neg_a=*/false, a, /*neg_b=*/false, b,
      /*c_mod=*/(short)0, c, /*reuse_a=*/false, /*reuse_b=*/false);
}

// ---------------------------------------------------------------------------
// GEMM:  out[M,N] = A[M,K] @ W[N,K]^T (+ bias[N])
// One wave per 32x32 output block (2x2 WMMA tiles): per k-step 2 A + 2 B
// fragment loads feed 4 WMMAs (8 bytes/lane per WMMA instead of 16).
// A fragment (32-bit 16x4): lane<16 -> K={k0,k0+1}; lane>=16 -> K={k0+2,k0+3}
// ---------------------------------------------------------------------------
__global__ __launch_bounds__(256) void gemm_awt_kernel(
    const float* __restrict__ A, const float* __restrict__ W,
    const float* __restrict__ bias, float* __restrict__ out,
    int M, int N, int K) {
  int wid  = (int)((blockIdx.x * blockDim.x + threadIdx.x) >> 5);
  int lane = (int)(threadIdx.x & 31);
  int tilesN = N >> 5;                 // 32-wide blocks
  int m0 = (wid / tilesN) << 5;
  int n0 = (wid % tilesN) << 5;
  int lm = lane & 15;
  int hi = lane >> 4;

  const float* a0 = A + (size_t)(m0 + lm) * K + 2 * hi;
  const float* a1 = a0 + (size_t)16 * K;
  const float* w0 = W + (size_t)(n0 + lm) * K + 2 * hi;
  const float* w1 = w0 + (size_t)16 * K;

  v8f acc00 = {}, acc01 = {}, acc10 = {}, acc11 = {};
#pragma unroll 4
  for (int k0 = 0; k0 < K; k0 += 4) {
    v2f fa0 = *(const v2f*)(a0 + k0);
    v2f fa1 = *(const v2f*)(a1 + k0);
    v2f fb0 = *(const v2f*)(w0 + k0);
    v2f fb1 = *(const v2f*)(w1 + k0);
    acc00 = wmma_f32_k4(fa0, fb0, acc00);
    acc01 = wmma_f32_k4(fa0, fb1, acc01);
    acc10 = wmma_f32_k4(fa1, fb0, acc10);
    acc11 = wmma_f32_k4(fa1, fb1, acc11);
  }
  float bb0 = bias ? bias[n0 + lm] : 0.0f;
  float bb1 = bias ? bias[n0 + 16 + lm] : 0.0f;
  float* o0 = out + (size_t)(m0 + 8 * hi) * N + n0 + lm;
  float* o1 = out + (size_t)(m0 + 16 + 8 * hi) * N + n0 + lm;
#pragma unroll
  for (int r = 0; r < 8; ++r) {
    o0[(size_t)r * N]      = acc00[r] + bb0;
    o0[(size_t)r * N + 16] = acc01[r] + bb1;
    o1[(size_t)r * N]      = acc10[r] + bb0;
    o1[(size_t)r * N + 16] = acc11[r] + bb1;
  }
}

// ---------------------------------------------------------------------------
// align[b,t,s] = sum_d v[d] * tanh(wq[b,t,d] + uh[b,s,d]); masked fill -1e9
// One block per (b,t) row; wq row + v staged in LDS; wave shuffle reduction.
// ---------------------------------------------------------------------------
__global__ __launch_bounds__(256) void align_kernel(
    const float* __restrict__ WQ, const float* __restrict__ UH,
    const float* __restrict__ v, const int* __restrict__ mask,
    float* __restrict__ align) {
  __shared__ __align__(16) float s_wq[DD];
  __shared__ __align__(16) float s_v[DD];
  int row = (int)blockIdx.x;  // b*T + t
  int b = row >> 6;
  int tid = (int)threadIdx.x;

  s_wq[tid]       = WQ[(size_t)row * DD + tid];
  s_wq[tid + 256] = WQ[(size_t)row * DD + tid + 256];
  s_v[tid]        = v[tid];
  s_v[tid + 256]  = v[tid + 256];
  __syncthreads();

  int w = tid >> 5, lane = tid & 31;
#pragma unroll 1
  for (int i = 0; i < 32; ++i) {
    int s = w + (i << 3);  // covers 0..255, 32 values per wave
    const float* uhr = UH + (size_t)(b * SS + s) * DD;
    float part = 0.0f;
#pragma unroll
    for (int j = 0; j < 4; ++j) {
      int d = j * 128 + lane * 4;
      float4 u  = *(const float4*)(uhr + d);
      float4 q  = *(const float4*)(s_wq + d);
      float4 vv = *(const float4*)(s_v + d);
      part += vv.x * tanhf(q.x + u.x);
      part += vv.y * tanhf(q.y + u.y);
      part += vv.z * tanhf(q.z + u.z);
      part += vv.w * tanhf(q.w + u.w);
    }
#pragma unroll
    for (int off = 16; off > 0; off >>= 1) part += __shfl_xor(part, off, 32);
    if (lane == 0) {
      float val = mask[b * SS + s] ? part : -1e9f;
      align[(size_t)row * SS + s] = val;
    }
  }
}

// ---------------------------------------------------------------------------
// sparsemax over S=256. One block (256 thr) per (b,t) row.
// max-reduce -> bitonic sort desc -> inclusive scan -> support -> tau -> clip
// Writes P (B,T,S) scratch and the (T,B,S) output region.
// ---------------------------------------------------------------------------
__global__ __launch_bounds__(256) void sparsemax_kernel(
    const float* __restrict__ align, float* __restrict__ P,
    float* __restrict__ out_align) {
  __shared__ float s_z[SS];
  __shared__ float s_srt[SS];
  __shared__ float s_cs[SS];
  __shared__ float s_red[SS];
  int row = (int)blockIdx.x;  // b*T + t
  int b = row >> 6, t = row & 63;
  int tid = (int)threadIdx.x;

  float z = align[(size_t)row * SS + tid];

  // row max
  s_red[tid] = z;
  __syncthreads();
  for (int off = 128; off > 0; off >>= 1) {
    if (tid < off) s_red[tid] = fmaxf(s_red[tid], s_red[tid + off]);
    __syncthreads();
  }
  z -= s_red[0];
  __syncthreads();
  s_z[tid] = z;
  s_srt[tid] = z;
  __syncthreads();

  // bitonic sort, descending
  for (int k = 2; k <= 256; k <<= 1) {
    for (int j = k >> 1; j > 0; j >>= 1) {
      int ixj = tid ^ j;
      if (ixj > tid) {
        float a = s_srt[tid], c = s_srt[ixj];
        bool desc = ((tid & k) == 0);
        if (desc ? (a < c) : (a > c)) { s_srt[tid] = c; s_srt[ixj] = a; }
      }
      __syncthreads();
    }
  }

  // inclusive scan of sorted values
  s_cs[tid] = s_srt[tid];
  __syncthreads();
  for (int off = 1; off < 256; off <<= 1) {
    float add = (tid >= off) ? s_cs[tid - off] : 0.0f;
    __syncthreads();
    s_cs[tid] += add;
    __syncthreads();
  }

  // support size: (k * z_srt[k-1]) > (cumsum[k-1] - 1)
  float flag = (((float)(tid + 1) * s_srt[tid]) > (s_cs[tid] - 1.0f)) ? 1.0f : 0.0f;
  s_red[tid] = flag;
  __syncthreads();
  for (int off = 128; off > 0; off >>= 1) {
    if (tid < off) s_red[tid] += s_red[tid + off];
    __syncthreads();
  }
  int supp = (int)s_red[0];
  float tau = (s_cs[supp - 1] - 1.0f) / (float)supp;
  float p = fmaxf(s_z[tid] - tau, 0.0f);

  P[(size_t)row * SS + tid] = p;
  out_align[((size_t)t * BB + b) * SS + tid] = p;  // (T,B,S)
}

// ---------------------------------------------------------------------------
// c[b] = P[b](T x S) @ H[b](S x D). One wave per 32x32 block (2x2 tiles).
// B fragments gathered with stride D (memory_bank is row-major over d).
// ---------------------------------------------------------------------------
__global__ __launch_bounds__(256) void ctx_gemm_kernel(
    const float* __restrict__ P, const float* __restrict__ H,
    float* __restrict__ C) {
  int wid  = (int)((blockIdx.x * blockDim.x + threadIdx.x) >> 5);
  int lane = (int)(threadIdx.x & 31);
  const int tilesN = DD / 32;          // 16
  const int tilesM = TT / 32;          // 2
  int b  = wid / (tilesM * tilesN);
  int tl = wid % (tilesM * tilesN);
  int m0 = (tl / tilesN) << 5;
  int n0 = (tl % tilesN) << 5;
  int lm = lane & 15, hi = lane >> 4;

  const float* Pb = P + (size_t)b * TT * SS;
  const float* Hb = H + (size_t)b * SS * DD;
  const float* a0 = Pb + (size_t)(m0 + lm) * SS + 2 * hi;
  const float* a1 = a0 + (size_t)16 * SS;

  v8f acc00 = {}, acc01 = {}, acc10 = {}, acc11 = {};
#pragma unroll 4
  for (int k0 = 0; k0 < SS; k0 += 4) {
    v2f fa0 = *(const v2f*)(a0 + k0);
    v2f fa1 = *(const v2f*)(a1 + k0);
    const float* hb = Hb + (size_t)(k0 + 2 * hi) * DD + n0 + lm;
    v2f fb0, fb1;
    fb0.x = hb[0];
    fb0.y = hb[DD];
    fb1.x = hb[16];
    fb1.y = hb[DD + 16];
    acc00 = wmma_f32_k4(fa0, fb0, acc00);
    acc01 = wmma_f32_k4(fa0, fb1, acc01);
    acc10 = wmma_f32_k4(fa1, fb0, acc10);
    acc11 = wmma_f32_k4(fa1, fb1, acc11);
  }
  float* Cb = C + (size_t)b * TT * DD;
  float* o0 = Cb + (size_t)(m0 + 8 * hi) * DD + n0 + lm;
  float* o1 = Cb + (size_t)(m0 + 16 + 8 * hi) * DD + n0 + lm;
#pragma unroll
  for (int r = 0; r < 8; ++r) {
    o0[(size_t)r * DD]      = acc00[r];
    o0[(size_t)r * DD + 16] = acc01[r];
    o1[(size_t)r * DD]      = acc10[r];
    o1[(size_t)r * DD + 16] = acc11[r];
  }
}

// ---------------------------------------------------------------------------
// attn[t,b,:] = [c ; src][b,t,:] @ W_out^T + b_out  (M=1024, K=1024, N=512)
// K split: first 512 from c, last 512 from source. 2x2 tile block per wave.
// Output written directly in (T,B,D) layout.
// ---------------------------------------------------------------------------
__global__ __launch_bounds__(256) void out_gemm_kernel(
    const float* __restrict__ Cc, const float* __restrict__ Src,
    const float* __restrict__ Wout, const float* __restrict__ bout,
    float* __restrict__ Out) {
  int wid  = (int)((blockIdx.x * blockDim.x + threadIdx.x) >> 5);
  int lane = (int)(threadIdx.x & 31);
  const int tilesN = DD / 32;  // 16
  int m0 = (wid / tilesN) << 5;
  int n0 = (wid % tilesN) << 5;
  int lm = lane & 15, hi = lane >> 4;

  const float* w0 = Wout + (size_t)(n0 + lm) * (2 * DD) + 2 * hi;
  const float* w1 = w0 + (size_t)16 * (2 * DD);
  const float* c0 = Cc  + (size_t)(m0 + lm) * DD + 2 * hi;
  const float* c1 = c0 + (size_t)16 * DD;
  const float* s0 = Src + (size_t)(m0 + lm) * DD + 2 * hi;
  const float* s1 = s0 + (size_t)16 * DD;

  v8f acc00 = {}, acc01 = {}, acc10 = {}, acc11 = {};
#pragma unroll 4
  for (int k0 = 0; k0 < DD; k0 += 4) {
    v2f fa0 = *(const v2f*)(c0 + k0);
    v2f fa1 = *(const v2f*)(c1 + k0);
    v2f fb0 = *(const v2f*)(w0 + k0);
    v2f fb1 = *(const v2f*)(w1 + k0);
    acc00 = wmma_f32_k4(fa0, fb0, acc00);
    acc01 = wmma_f32_k4(fa0, fb1, acc01);
    acc10 = wmma_f32_k4(fa1, fb0, acc10);
    acc11 = wmma_f32_k4(fa1, fb1, acc11);
  }
#pragma unroll 4
  for (int k0 = 0; k0 < DD; k0 += 4) {
    v2f fa0 = *(const v2f*)(s0 + k0);
    v2f fa1 = *(const v2f*)(s1 + k0);
    v2f fb0 = *(const v2f*)(w0 + DD + k0);
    v2f fb1 = *(const v2f*)(w1 + DD + k0);
    acc00 = wmma_f32_k4(fa0, fb0, acc00);
    acc01 = wmma_f32_k4(fa0, fb1, acc01);
    acc10 = wmma_f32_k4(fa1, fb0, acc10);
    acc11 = wmma_f32_k4(fa1, fb1, acc11);
  }
  float bb0 = bout[n0 + lm];
  float bb1 = bout[n0 + 16 + lm];
#pragma unroll
  for (int r = 0; r < 8; ++r) {
    int m = m0 + r + 8 * hi;        // m = b*T + t
    int t = m & 63, b = m >> 6;
    float* o = Out + ((size_t)t * BB + b) * DD + n0 + lm;
    o[0]  = acc00[r] + bb0;
    o[16] = acc01[r] + bb1;
    int m2 = m + 16;
    int t2 = m2 & 63, b2 = m2 >> 6;
    float* o2 = Out + ((size_t)t2 * BB + b2) * DD + n0 + lm;
    o2[0]  = acc10[r] + bb0;
    o2[16] = acc11[r] + bb1;
  }
}

// ---------------------------------------------------------------------------
extern "C" void kernel_launch(void* const* d_in, const int* in_sizes, int n_in,
                              void* d_out, int out_size, void* d_ws, size_t ws_size,
                              hipStream_t stream) {
  (void)in_sizes; (void)n_in; (void)out_size; (void)ws_size;
  const float* source  = (const float*)d_in[0];  // (B,T,D)
  const float* membank = (const float*)d_in[1];  // (B,S,D)
  const int*   mask    = (const int*)d_in[2];    // (B,S)
  const float* W_q     = (const float*)d_in[3];  // (D,D)
  const float* b_q     = (const float*)d_in[4];  // (D,)
  const float* W_c     = (const float*)d_in[5];  // (D,D)
  const float* v       = (const float*)d_in[6];  // (D,)
  const float* W_out   = (const float*)d_in[7];  // (D,2D)
  const float* b_out   = (const float*)d_in[8];  // (D,)

  float* out       = (float*)d_out;
  float* attn_out  = out;                         // (T,B,D) = 524288 floats
  float* align_out = out + (size_t)TT * BB * DD;  // (T,B,S) = 262144 floats

  float* ws    = (float*)d_ws;
  float* wq_ws = ws;                                   // (B*T, D)   524288
  float* uh_ws = wq_ws + (size_t)BB * TT * DD;         // (B*S, D)  2097152
  float* al_ws = uh_ws + (size_t)BB * SS * DD;         // (B*T, S)   262144
  float* p_ws  = al_ws + (size_t)BB * TT * SS;         // (B*T, S)   262144
  float* c_ws  = p_ws  + (size_t)BB * TT * SS;         // (B*T, D)   524288

  // 1) wq = source @ W_q^T + b_q : M=1024,N=512,K=512 -> 512 waves (32x32 ea)
  gemm_awt_kernel<<<64, 256, 0, stream>>>(source, W_q, b_q, wq_ws,
                                          BB * TT, DD, DD);
  // 2) uh = memory_bank @ W_c^T  : M=4096,N=512,K=512 -> 2048 waves
  gemm_awt_kernel<<<256, 256, 0, stream>>>(membank, W_c, nullptr, uh_ws,
                                           BB * SS, DD, DD);
  // 3) align scores (tanh MLP + mask) : 1024 blocks x 8 waves
  align_kernel<<<BB * TT, 256, 0, stream>>>(wq_ws, uh_ws, v, mask, al_ws);
  // 4) sparsemax rows -> P scratch + (T,B,S) output
  sparsemax_kernel<<<BB * TT, 256, 0, stream>>>(al_ws, p_ws, align_out);
  // 5) c = P @ memory_bank per batch: 16*(2*16)=512 waves
  ctx_gemm_kernel<<<64, 256, 0, stream>>>(p_ws, membank, c_ws);
  // 6) attn_h = [c;src] @ W_out^T + b_out, transposed store (T,B,D)
  out_gemm_kernel<<<64, 256, 0, stream>>>(c_ws, source, W_out, b_out, attn_out);
}